// AverageDistanceLoss_15839839388141
// MI455X (gfx1250) — compile-verified
//
#include <hip/hip_runtime.h>
#include <hip/hip_bf16.h>

#define NCLS   22
#define NPTS   1024
#define MARGIN 0.01f

typedef __attribute__((ext_vector_type(2))) float v2f;
typedef __attribute__((ext_vector_type(8))) float v8f;

__device__ __forceinline__ void quat_to_rot(const float q[4], float R[9]) {
    const float w = q[0], x = q[1], y = q[2], z = q[3];
    R[0] = 1.f - 2.f * (y * y + z * z); R[1] = 2.f * (x * y - z * w); R[2] = 2.f * (x * z + y * w);
    R[3] = 2.f * (x * y + z * w); R[4] = 1.f - 2.f * (x * x + z * z); R[5] = 2.f * (y * z - x * w);
    R[6] = 2.f * (x * z - y * w); R[7] = 2.f * (y * z + x * w); R[8] = 1.f - 2.f * (x * x + y * y);
}

// min with a DPP16-permuted copy of x (permutation stays within each 16-lane row).
// CTRL must be a compile-time constant for the builtin -> template parameter.
template <int CTRL>
__device__ __forceinline__ float min_dpp16(float x) {
    int xi = __float_as_int(x);
    int yi = __builtin_amdgcn_update_dpp(xi, xi, CTRL, 0xF, 0xF, true);
    return fminf(x, __int_as_float(yi));
}

// One block per batch element. 256 threads = 8 waves (wave32).
__global__ __launch_bounds__(256) void adl_batch_kernel(
    const float* __restrict__ poses_pred,
    const float* __restrict__ poses_target,
    const float* __restrict__ poses_weight,
    const float* __restrict__ points,
    const float* __restrict__ symmetry,
    float* __restrict__ block_out)
{
    // Packed WMMA operands staged in LDS:
    //   sA[p] = (-2*xp, 1)    -> A rows (16x4 f32 tiles)
    //   sB[q] = (xg, |xg|^2)  -> B cols (4x16 f32 tiles)
    //   sNxp[p] = |xp|^2      -> C (broadcast along N)
    __shared__ float sA[NPTS][4];
    __shared__ float sB[NPTS][4];
    __shared__ float sNxp[NPTS];
    __shared__ float sPartial[8];

    const int b    = blockIdx.x;
    const int tid  = threadIdx.x;
    const int lane = tid & 31;
    const int wave = tid >> 5;

    // Class selection: first i with weight[b, 4*i] > 0 (block-uniform).
    int cls = -1;
    #pragma unroll
    for (int i = 0; i < NCLS; ++i) {
        if (cls < 0 && poses_weight[b * 4 * NCLS + 4 * i] > 0.f) cls = i;
    }
    if (cls < 0) {                 // invalid sample contributes 0
        if (tid == 0) block_out[b] = 0.f;
        return;
    }
    const bool sym = symmetry[cls] > 0.f;

    float qp[4], qg[4];
    #pragma unroll
    for (int k = 0; k < 4; ++k) {
        qp[k] = poses_pred  [b * 4 * NCLS + 4 * cls + k];
        qg[k] = poses_target[b * 4 * NCLS + 4 * cls + k];
    }
    float Rp[9], Rg[9];
    quat_to_rot(qp, Rp);
    quat_to_rot(qg, Rg);

    // Rotate points, pack WMMA operands into LDS, accumulate diagonal (ADD) loss.
    float diag_sum = 0.f;
    for (int p = tid; p < NPTS; p += 256) {
        const float* pt3 = &points[((size_t)cls * NPTS + p) * 3];
        const float x = pt3[0], y = pt3[1], z = pt3[2];
        const float ax = Rp[0] * x + Rp[1] * y + Rp[2] * z;
        const float ay = Rp[3] * x + Rp[4] * y + Rp[5] * z;
        const float az = Rp[6] * x + Rp[7] * y + Rp[8] * z;
        const float gx = Rg[0] * x + Rg[1] * y + Rg[2] * z;
        const float gy = Rg[3] * x + Rg[4] * y + Rg[5] * z;
        const float gz = Rg[6] * x + Rg[7] * y + Rg[8] * z;
        sA[p][0] = -2.f * ax; sA[p][1] = -2.f * ay; sA[p][2] = -2.f * az; sA[p][3] = 1.f;
        sB[p][0] = gx; sB[p][1] = gy; sB[p][2] = gz;
        sB[p][3] = gx * gx + gy * gy + gz * gz;
        sNxp[p]  = ax * ax + ay * ay + az * az;
        const float dx = ax - gx, dy = ay - gy, dz = az - gz;
        const float dd = dx * dx + dy * dy + dz * dz;
        diag_sum += 0.5f * fmaxf(dd - MARGIN, 0.f);
    }

    float wave_sum;
    if (!sym) {
        // ADD path: plain wave reduction of diagonal loss.
        wave_sum = diag_sum;
        #pragma unroll
        for (int off = 16; off >= 1; off >>= 1)
            wave_sum += __shfl_xor(wave_sum, off, 32);
    } else {
        // ADD-S path: D = A x B + C gives d2[p][q] exactly, one WMMA per 16x16 tile.
        // Note: min_q(max(d2,0)) == max(min_q(d2),0), and the hinge absorbs the
        // clamp (m<0 -> 0 either way), so no per-element clamp in the hot loop.
        __syncthreads();
        const int half = lane >> 4;   // 0: K=0,1 / M=r ; 1: K=2,3 / M=r+8
        const int lm   = lane & 15;
        wave_sum = 0.f;

        for (int ti = wave; ti < NPTS / 16; ti += 8) {
            const int rowbase = ti * 16;
            // A 16x4 f32: lanes 0-15 -> M=0..15, K=0(v0)/1(v1); lanes 16-31 -> K=2/3.
            v2f A;
            A.x = sA[rowbase + lm][half * 2];
            A.y = sA[rowbase + lm][half * 2 + 1];
            // C 16x16 f32: VGPR r -> row M = r + half*8, value |xp_M|^2.
            v8f Cv;
            #pragma unroll
            for (int r = 0; r < 8; ++r)
                Cv[r] = sNxp[rowbase + half * 8 + r];

            v8f rowmin;
            #pragma unroll
            for (int r = 0; r < 8; ++r) rowmin[r] = 3.0e38f;

            // Software-pipelined sweep over 64 B tiles: prefetch tile tj+1
            // while the WMMA + min chain consumes tile tj.
            v2f Bv;
            Bv.x = sB[lm][half * 2];
            Bv.y = sB[lm][half * 2 + 1];
            #pragma unroll 4
            for (int tj = 0; tj < NPTS / 16 - 1; ++tj) {
                const int cb = (tj + 1) * 16;
                v2f Bn;
                Bn.x = sB[cb + lm][half * 2];
                Bn.y = sB[cb + lm][half * 2 + 1];
                v8f D = __builtin_amdgcn_wmma_f32_16x16x4_f32(
                    false, A, false, Bv, (short)0, Cv, false, false);
                #pragma unroll
                for (int r = 0; r < 8; ++r)
                    rowmin[r] = fminf(rowmin[r], D[r]);
                Bv = Bn;
            }
            {   // epilogue tile
                v8f D = __builtin_amdgcn_wmma_f32_16x16x4_f32(
                    false, A, false, Bv, (short)0, Cv, false, false);
                #pragma unroll
                for (int r = 0; r < 8; ++r)
                    rowmin[r] = fminf(rowmin[r], D[r]);
            }

            // Column-min across the 16 N-lanes of each half via DPP16
            // (quad_perm xor1, xor2, then row_ror:4, row_ror:8 — rotations
            // are valid reduction permutations; DPP16 rows == lane halves).
            #pragma unroll
            for (int r = 0; r < 8; ++r) {
                float m = rowmin[r];
                m = min_dpp16<0xB1>(m);   // quad_perm(1,0,3,2)  : xor 1
                m = min_dpp16<0x4E>(m);   // quad_perm(2,3,0,1)  : xor 2
                m = min_dpp16<0x124>(m);  // row_ror:4
                m = min_dpp16<0x128>(m);  // row_ror:8
                if (lm == 0)
                    wave_sum += 0.5f * fmaxf(m - MARGIN, 0.f);
            }
        }
        // Combine the two half-wave contributions (lanes 0 and 16).
        wave_sum += __shfl_xor(wave_sum, 16, 32);
    }

    if (lane == 0) sPartial[wave] = wave_sum;
    __syncthreads();
    if (tid == 0) {
        float s = 0.f;
        #pragma unroll
        for (int w = 0; w < 8; ++w) s += sPartial[w];
        block_out[b] = s;
    }
}

// Deterministic final reduction (no float atomics -> stable across graph replays).
__global__ __launch_bounds__(128) void adl_reduce_kernel(
    const float* __restrict__ block_out, float* __restrict__ out,
    int nblocks, float scale)
{
    __shared__ float s[128];
    const int tid = threadIdx.x;
    float v = 0.f;
    for (int i = tid; i < nblocks; i += 128) v += block_out[i];
    s[tid] = v;
    __syncthreads();
    #pragma unroll
    for (int off = 64; off > 0; off >>= 1) {
        if (tid < off) s[tid] += s[tid + off];
        __syncthreads();
    }
    if (tid == 0) out[0] = s[0] * scale;
}

extern "C" void kernel_launch(void* const* d_in, const int* in_sizes, int n_in,
                              void* d_out, int out_size, void* d_ws, size_t ws_size,
                              hipStream_t stream) {
    const float* poses_pred   = (const float*)d_in[0];
    const float* poses_target = (const float*)d_in[1];
    const float* poses_weight = (const float*)d_in[2];
    const float* points       = (const float*)d_in[3];
    const float* symmetry     = (const float*)d_in[4];

    const int B = in_sizes[0] / (4 * NCLS);
    float* block_out = (float*)d_ws;              // B floats of scratch
    const float scale = 1.0f / (float)(B * NPTS);

    adl_batch_kernel<<<B, 256, 0, stream>>>(
        poses_pred, poses_target, poses_weight, points, symmetry, block_out);
    adl_reduce_kernel<<<1, 128, 0, stream>>>(block_out, (float*)d_out, B, scale);
}